// BahdanauAttention_17257178595902
// MI455X (gfx1250) — compile-verified
//
#include <hip/hip_runtime.h>

typedef __attribute__((ext_vector_type(2))) float v2f;
typedef __attribute__((ext_vector_type(8))) float v8f;

#define B_   128
#define TX_  400
#define ENC_ 512

// One wave (wave32) per (batch b, 16-wide e-block) tile.
// Column sums over t via V_WMMA_F32_16X16X4_F32 with A = ones(16x4):
//   D[m,n] += sum_k B[k,n]  -> every row of D holds the per-column sums.
// 4 independent accumulators break the D->C dependency chain.
__global__ __launch_bounds__(128) void bahdanau_ctx_sum_wmma(
    const float* __restrict__ a, float* __restrict__ ctx)
{
    const int wave = threadIdx.x >> 5;
    const int lane = threadIdx.x & 31;
    const int tile = blockIdx.x * 4 + wave;   // 0 .. 4095
    const int b    = tile >> 5;               // / (ENC/16 = 32)
    const int eb   = tile & 31;
    const int n    = lane & 15;               // column within e-block
    const int half = lane >> 4;               // which k-pair this lane supplies

    const float* __restrict__ base =
        a + (size_t)b * (TX_ * ENC_) + (size_t)eb * 16 + n;

    const v2f av = {1.0f, 1.0f};              // A = ones
    v8f acc0 = {}, acc1 = {}, acc2 = {}, acc3 = {};

    // 400 = 25 * 16 : four K=4 WMMAs per iteration, no tail.
    for (int t0 = 0; t0 < TX_; t0 += 16) {
        v2f b0, b1, b2, b3;
        const float* p = base + (size_t)t0 * ENC_ + (size_t)(2 * half) * ENC_;
        b0.x = p[0 * (size_t)ENC_];  b0.y = p[1 * (size_t)ENC_];
        b1.x = p[4 * (size_t)ENC_];  b1.y = p[5 * (size_t)ENC_];
        b2.x = p[8 * (size_t)ENC_];  b2.y = p[9 * (size_t)ENC_];
        b3.x = p[12 * (size_t)ENC_]; b3.y = p[13 * (size_t)ENC_];

        acc0 = __builtin_amdgcn_wmma_f32_16x16x4_f32(
            false, av, false, b0, (short)0, acc0, false, false);
        acc1 = __builtin_amdgcn_wmma_f32_16x16x4_f32(
            false, av, false, b1, (short)0, acc1, false, false);
        acc2 = __builtin_amdgcn_wmma_f32_16x16x4_f32(
            false, av, false, b2, (short)0, acc2, false, false);
        acc3 = __builtin_amdgcn_wmma_f32_16x16x4_f32(
            false, av, false, b3, (short)0, acc3, false, false);
    }

    // VGPR0, lanes 0..15 hold D[0][n] = column sum for e0+n.
    float s = acc0[0] + acc1[0] + acc2[0] + acc3[0];
    if (half == 0) {
        ctx[(size_t)b * ENC_ + (size_t)eb * 16 + n] = s;
    }
}

// softmax over a size-1 axis => alpha is identically 1.0
__global__ __launch_bounds__(256) void bahdanau_alpha_ones(
    float* __restrict__ alpha, int nelem)
{
    int i = blockIdx.x * blockDim.x + threadIdx.x;
    if (i < nelem) alpha[i] = 1.0f;
}

extern "C" void kernel_launch(void* const* d_in, const int* in_sizes, int n_in,
                              void* d_out, int out_size, void* d_ws, size_t ws_size,
                              hipStream_t stream)
{
    const float* a = (const float*)d_in[0];   // (B, TX, ENC) f32

    float* out   = (float*)d_out;
    float* ctx   = out;                       // (B, 1, ENC) = 65536 floats
    float* alpha = out + (size_t)B_ * ENC_;   // (B, TX)     = 51200 floats

    // 4096 tiles, 4 waves (128 threads) per block -> 1024 blocks
    bahdanau_ctx_sum_wmma<<<(B_ * (ENC_ / 16)) / 4, 128, 0, stream>>>(a, ctx);

    const int n_alpha = B_ * TX_;
    bahdanau_alpha_ones<<<(n_alpha + 255) / 256, 256, 0, stream>>>(alpha, n_alpha);
}